// Encoder_22024592294208
// MI455X (gfx1250) — compile-verified
//
#include <hip/hip_runtime.h>
#include <hip/hip_bf16.h>
#include <math.h>

#define LSEQ   2048
#define BATCH  8
#define DMODEL 512
#define TI     16

typedef float v2f __attribute__((ext_vector_type(2)));
typedef float v4f __attribute__((ext_vector_type(4)));
typedef float v8f __attribute__((ext_vector_type(8)));

// d_out layout: scores [B,L,L] | embedding [B,L,512] | t_diff [B,L,L]
#define EMB_OFF ((size_t)BATCH * LSEQ * LSEQ)                    // 33554432
#define TD_OFF  (EMB_OFF + (size_t)BATCH * LSEQ * DMODEL)        // 41943040

__launch_bounds__(256)
__global__ void scores_tdiff_kernel(const float* __restrict__ et,
                                    float* __restrict__ out) {
  __shared__ float sh_t[LSEQ];        // event_time[b, :]
  __shared__ float sh_part[8][16];    // per-wave row-sum partials
  __shared__ float sh_inv[TI];        // 1 / softmax denominator per row

  const int b   = blockIdx.y;
  const int i0  = blockIdx.x * TI;
  const int tid = threadIdx.x;

  // ---- Stage event_time[b, 0..2048) (8 KB) into LDS via CDNA5 async copy ----
  {
    const float* src = et + (size_t)b * LSEQ;
    unsigned lds0 = (unsigned)(size_t)(&sh_t[tid * 4]);
    unsigned off0 = (unsigned)(tid * 16);
    asm volatile("global_load_async_to_lds_b128 %0, %1, %2"
                 :: "v"(lds0), "v"(off0), "s"(src) : "memory");
    unsigned lds1 = (unsigned)(size_t)(&sh_t[1024 + tid * 4]);
    unsigned off1 = (unsigned)(4096 + tid * 16);
    asm volatile("global_load_async_to_lds_b128 %0, %1, %2"
                 :: "v"(lds1), "v"(off1), "s"(src) : "memory");
    asm volatile("s_wait_asynccnt 0" ::: "memory");
  }
  __syncthreads();

  const int w = tid >> 5;        // wave id 0..7
  const int l = tid & 31;        // lane
  const int r = l & 15;          // row within 16-row tile
  const int h = l >> 4;          // half-wave (selects K=0,1 vs K=2,3 slots)
  const int imax = i0 + TI - 1;
  const int ig   = i0 + r;       // this lane's global row index
  const float ti   = sh_t[ig];
  const float kinv = 1.0f / 200.0f;   // 1 / T_MAX

  // ---- Pass 1: softmax denominators. Lane accumulates its column slice of
  // exp(score - 1) into the two f32 A-matrix K-slots it owns; columns are
  // interleaved j = w + 8*slot + 32*m so the causal break balances waves. ----
  float a0 = 0.0f, a1 = 0.0f;
  for (int m = 0; m < LSEQ / 32; ++m) {
    const int jb = w + 32 * m;          // wave-uniform
    if (jb > imax) break;               // tile fully above diagonal: skip
    const int jA = jb + 16 * h;
    const int jB = jA + 8;
    const float dA = (ti - sh_t[jA]) * kinv;
    const float dB = (ti - sh_t[jB]) * kinv;
    const float sA = expf(-0.5f * dA * dA);   // kernel value (<= 1, max @ j=i)
    const float sB = expf(-0.5f * dB * dB);
    a0 += (jA <= ig) ? expf(sA - 1.0f) : 0.0f; // row max is exactly 1.0
    a1 += (jB <= ig) ? expf(sB - 1.0f) : 0.0f;
  }

  // One f32 WMMA with B = ones collapses the 4 per-row accumulators:
  // D[i,n] = sum_k A[i,k]  (row sum replicated across all 16 columns).
  v2f a; a[0] = a0; a[1] = a1;
  v2f bones; bones[0] = 1.0f; bones[1] = 1.0f;
  v8f c = {};
  c = __builtin_amdgcn_wmma_f32_16x16x4_f32(false, a, false, bones,
                                            (short)0, c, false, false);

  // C/D layout: lane<16 -> c[v] = D[M=v, N=lane]; lane>=16 -> D[M=v+8, ...]
  if (l == 0) {
#pragma unroll
    for (int v = 0; v < 8; ++v) sh_part[w][v] = c[v];
  } else if (l == 16) {
#pragma unroll
    for (int v = 0; v < 8; ++v) sh_part[w][8 + v] = c[v];
  }
  __syncthreads();
  if (tid < TI) {
    float s = 0.0f;
#pragma unroll
    for (int ww = 0; ww < 8; ++ww) s += sh_part[ww][tid];
    sh_inv[tid] = 1.0f / s;
  }
  __syncthreads();

  // ---- Pass 2: stream t_diff (full) and scores (0 above diagonal) ----
  // 16 rows x 2048 cols = 8192 float4 chunks; 32 per thread, coalesced.
  for (int it = 0; it < 32; ++it) {
    const int idx  = tid + 256 * it;
    const int row  = idx >> 9;              // 512 chunks per row
    const int col  = (idx & 511) << 2;
    const int irow = i0 + row;
    const float tir = sh_t[irow];
    const float inv = sh_inv[row];
    v4f tdv, scv;
    if (col > irow) {                       // fully masked chunk: no exp work
#pragma unroll
      for (int cc = 0; cc < 4; ++cc) { tdv[cc] = tir - sh_t[col + cc]; scv[cc] = 0.0f; }
    } else {
#pragma unroll
      for (int cc = 0; cc < 4; ++cc) {
        const int j   = col + cc;
        const float dt = tir - sh_t[j];
        tdv[cc] = dt;
        const float d = dt * kinv;
        const float s = expf(-0.5f * d * d);
        scv[cc] = (j <= irow) ? expf(s - 1.0f) * inv : 0.0f;
      }
    }
    const size_t off = (((size_t)(b * LSEQ + irow)) << 11) + (size_t)col;
    __builtin_nontemporal_store(scv, (v4f*)(out + off));
    __builtin_nontemporal_store(tdv, (v4f*)(out + TD_OFF + off));
  }
}

__launch_bounds__(256)
__global__ void embedding_kernel(const float* __restrict__ et,
                                 float* __restrict__ out) {
  __shared__ float winv[DMODEL];   // 1 / w_k
  __shared__ float tt[TI];

  const int b   = blockIdx.y;
  const int i0  = blockIdx.x * TI;
  const int tid = threadIdx.x;

#pragma unroll
  for (int q = 0; q < 2; ++q) {
    const int k = tid + 256 * q;
    const float e = (float)(2 * (k / 2 + 1)) / (float)DMODEL;
    winv[k] = 1.0f / powf(10000.0f, e);
  }
  if (tid < TI) tt[tid] = et[(size_t)b * LSEQ + i0 + tid];
  __syncthreads();

  float* emb = out + EMB_OFF;
  // 16 rows x 512 = 2048 float4 chunks; 8 per thread.
  for (int it = 0; it < 8; ++it) {
    const int idx = tid + 256 * it;
    const int row = idx >> 7;                // 128 chunks per row
    const int kc  = (idx & 127) << 2;
    const float t = tt[row];
    v4f v;
#pragma unroll
    for (int cc = 0; cc < 4; ++cc) {
      const int k = kc + cc;
      const float ph = t * winv[k];
      v[cc] = (k & 1) ? cosf(ph) : sinf(ph); // even k -> sin, odd -> cos
    }
    const size_t off = (((size_t)(b * LSEQ + i0 + row)) << 9) + (size_t)kc;
    __builtin_nontemporal_store(v, (v4f*)(emb + off));
  }
}

extern "C" void kernel_launch(void* const* d_in, const int* in_sizes, int n_in,
                              void* d_out, int out_size, void* d_ws, size_t ws_size,
                              hipStream_t stream) {
  (void)in_sizes; (void)n_in; (void)out_size; (void)d_ws; (void)ws_size;
  // d_in[0] = event_type (unused by outputs), d_in[1] = event_time (f32),
  // d_in[2] = type_emb_table (unused by outputs)
  const float* event_time = (const float*)d_in[1];
  float* out = (float*)d_out;

  dim3 grid(LSEQ / TI, BATCH);
  scores_tdiff_kernel<<<grid, 256, 0, stream>>>(event_time, out);
  embedding_kernel<<<grid, 256, 0, stream>>>(event_time, out);
}